// BiGNNLayer_44616120271338
// MI455X (gfx1250) — compile-verified
//
#include <hip/hip_runtime.h>

#define DIM    256
#define DHALF  128
#define NVIEWS 2
#define LDSPAD 4   // row stride DIM+4: 16B-aligned rows, conflict-free b64 reads

typedef __attribute__((ext_vector_type(2))) float v2f;
typedef __attribute__((ext_vector_type(8))) float v8f;

// ---------------------------------------------------------------------------
// Zero fill (float4 main body + scalar tail).
// ---------------------------------------------------------------------------
__global__ void zero_f32(float* __restrict__ p, size_t n) {
  const size_t n4 = n >> 2;
  size_t i = (size_t)blockIdx.x * blockDim.x + threadIdx.x;
  const size_t stride = (size_t)gridDim.x * blockDim.x;
  float4* p4 = (float4*)p;
  const float4 z = make_float4(0.f, 0.f, 0.f, 0.f);
  for (size_t j = i; j < n4; j += stride) p4[j] = z;
  for (size_t j = (n4 << 2) + i; j < n; j += stride) p[j] = 0.0f;
}

// ---------------------------------------------------------------------------
// Edge scatter-aggregation: agg[dst] += x[src]; deg[dst] += 1.
// One wave per edge; each lane moves 8 channels (two b128 loads => one
// coalesced 1KB row read per wave) and issues 8 f32 global atomics.
// agg (51.2MB) is processed one direction at a time so it stays L2-resident.
// ---------------------------------------------------------------------------
__global__ void edge_agg(const float* __restrict__ x,
                         const int* __restrict__ src,
                         const int* __restrict__ dst,
                         float* __restrict__ agg,
                         float* __restrict__ deg, int E) {
  const int lane = threadIdx.x & 31;
  const int wave = blockIdx.x * (blockDim.x >> 5) + (threadIdx.x >> 5);
  const int nwaves = gridDim.x * (blockDim.x >> 5);
  for (int e = wave; e < E; e += nwaves) {
    const int s = src[e];
    const int d = dst[e];
    const float4* xr = (const float4*)(x + (size_t)s * DIM);
    const float4 a0 = xr[lane * 2 + 0];
    const float4 a1 = xr[lane * 2 + 1];
    float* ar = agg + (size_t)d * DIM + lane * 8;
    atomicAdd(ar + 0, a0.x); atomicAdd(ar + 1, a0.y);
    atomicAdd(ar + 2, a0.z); atomicAdd(ar + 3, a0.w);
    atomicAdd(ar + 4, a1.x); atomicAdd(ar + 5, a1.y);
    atomicAdd(ar + 6, a1.z); atomicAdd(ar + 7, a1.w);
    if (lane == 0) atomicAdd(deg + d, 1.0f);
  }
}

// ---------------------------------------------------------------------------
// h = relu((agg@W) * (1/max(deg,1)) + b); acc[:, col_off:col_off+128] += h
// (row scaling commutes with the GEMM, so deg division moves to the epilogue)
// One wave per 16-node tile. Raw agg tile staged into LDS with CDNA5
// GLOBAL_LOAD_ASYNC_TO_LDS_B128 (ASYNCcnt), then f32 WMMA 16x16x4 K-loop.
// A layout: lane<16: M=lane, K={k0,k0+1}; lane>=16: M=lane-16, K={k0+2,k0+3}.
// B mirrors; C/D: vgpr r -> row r / r+8, col = lane&15.
// ---------------------------------------------------------------------------
__global__ __launch_bounds__(32)
void gnn_gemm(const float* __restrict__ agg, const float* __restrict__ deg,
              const float* __restrict__ W,     // [DIM][DHALF]
              const float* __restrict__ bias,  // [DHALF]
              float* __restrict__ acc,         // [N][DIM]
              int col_off) {
  __shared__ float lds_a[16][DIM + LDSPAD];
  __shared__ float lds_rd[16];
  const int r0 = blockIdx.x * 16;
  const int lane = threadIdx.x;
  const int l15 = lane & 15;
  const int half = lane >> 4;

  // Async-DMA the raw 16x256 f32 tile into LDS: 1024 b128 transfers.
  const unsigned lds_base = (unsigned)(uintptr_t)(&lds_a[0][0]);
  for (int i = lane; i < 16 * (DIM / 4); i += 32) {
    const int r = i >> 6;       // row (64 float4 per row)
    const int c4 = i & 63;      // float4 within row
    const unsigned long long ga =
        (unsigned long long)(uintptr_t)(agg + (size_t)(r0 + r) * DIM + c4 * 4);
    const unsigned la = lds_base + (unsigned)(r * (DIM + LDSPAD) + c4 * 4) * 4u;
    asm volatile("global_load_async_to_lds_b128 %0, %1, off"
                 :: "v"(la), "v"(ga) : "memory");
  }
  if (lane < 16) {
    float dg = deg[r0 + lane];
    lds_rd[lane] = 1.0f / (dg > 1.0f ? dg : 1.0f);
  }
  asm volatile("s_wait_asynccnt 0x0" ::: "memory");
  __syncthreads();

  for (int nt = 0; nt < DHALF / 16; ++nt) {
    const int n = nt * 16 + l15;
    v8f c = {};
#pragma unroll 4
    for (int k0 = 0; k0 < DIM; k0 += 4) {
      const int ka = k0 + half * 2;
      v2f a, b;
      a.x = lds_a[l15][ka];
      a.y = lds_a[l15][ka + 1];
      b.x = W[(size_t)ka * DHALF + n];
      b.y = W[(size_t)(ka + 1) * DHALF + n];
      c = __builtin_amdgcn_wmma_f32_16x16x4_f32(false, a, false, b, (short)0, c,
                                                false, false);
    }
    const float bn = bias[n];
#pragma unroll
    for (int r = 0; r < 8; ++r) {
      const int m = r + half * 8;
      float h = c[r] * lds_rd[m] + bn;
      h = h > 0.0f ? h : 0.0f;
      float* p = acc + (size_t)(r0 + m) * DIM + col_off + nt * 16 + l15;
      *p += h;  // stream-ordered RMW across rounds; unique (m,n) per lane/vgpr
    }
  }
}

// ---------------------------------------------------------------------------
// feats = relu(acc) @ W1 + b1 + inps, in-place over d_out (acc).
// Relu is elementwise on A, so it's fused into a float4-wide staging pass
// (1024 vector ops) instead of per-fragment max in the K-loop (2048 ops).
// ---------------------------------------------------------------------------
__global__ __launch_bounds__(32)
void final_fuse(const float* __restrict__ W1,   // [DIM][DIM]
                const float* __restrict__ b1,   // [DIM]
                const float* __restrict__ inps, // [N][DIM]
                float* __restrict__ out) {      // acc in / feats out [N][DIM]
  __shared__ float lds_a[16][DIM + LDSPAD];
  const int r0 = blockIdx.x * 16;
  const int lane = threadIdx.x;
  const int l15 = lane & 15;
  const int half = lane >> 4;

  for (int i = lane; i < 16 * (DIM / 4); i += 32) {
    const int r = i >> 6;
    const int c4 = i & 63;
    float4 v = *(const float4*)(out + (size_t)(r0 + r) * DIM + c4 * 4);
    v.x = v.x > 0.0f ? v.x : 0.0f;
    v.y = v.y > 0.0f ? v.y : 0.0f;
    v.z = v.z > 0.0f ? v.z : 0.0f;
    v.w = v.w > 0.0f ? v.w : 0.0f;
    *(float4*)(&lds_a[r][c4 * 4]) = v;  // 16B-aligned (row stride 1040B)
  }
  __syncthreads();

  for (int nt = 0; nt < DIM / 16; ++nt) {
    const int n = nt * 16 + l15;
    v8f c = {};
#pragma unroll 4
    for (int k0 = 0; k0 < DIM; k0 += 4) {
      const int ka = k0 + half * 2;
      v2f a, b;
      a.x = lds_a[l15][ka];
      a.y = lds_a[l15][ka + 1];
      b.x = W1[(size_t)ka * DIM + n];
      b.y = W1[(size_t)(ka + 1) * DIM + n];
      c = __builtin_amdgcn_wmma_f32_16x16x4_f32(false, a, false, b, (short)0, c,
                                                false, false);
    }
    const float bn = b1[n];
#pragma unroll
    for (int r = 0; r < 8; ++r) {
      const int m = r + half * 8;
      const size_t idx = (size_t)(r0 + m) * DIM + n;
      out[idx] = c[r] + bn + inps[idx];
    }
  }
}

// ---------------------------------------------------------------------------
extern "C" void kernel_launch(void* const* d_in, const int* in_sizes, int n_in,
                              void* d_out, int out_size, void* d_ws, size_t ws_size,
                              hipStream_t stream) {
  const float* inps = (const float*)d_in[0];
  const int* fw_e   = (const int*)d_in[1];
  const int* bw_e   = (const int*)d_in[2];
  const float* W_fw = (const float*)d_in[3];
  const float* b_fw = (const float*)d_in[4];
  const float* W_bw = (const float*)d_in[5];
  const float* b_bw = (const float*)d_in[6];
  const float* W1   = (const float*)d_in[7];
  const float* b1   = (const float*)d_in[8];
  float* out = (float*)d_out;

  const int N = in_sizes[0] / DIM;          // 50000 (multiple of 16)
  const int E = in_sizes[1] / (NVIEWS * 2); // 800000
  const int ntiles = N / 16;

  float* agg = (float*)d_ws;                 // [N][DIM]
  float* deg = agg + (size_t)N * DIM;        // [N]  (contiguous after agg)

  // acc lives in d_out; zero it once per launch (deterministic).
  zero_f32<<<1024, 256, 0, stream>>>(out, (size_t)N * DIM);

  for (int v = 0; v < NVIEWS; ++v) {
    // backward direction -> cols [0, DHALF)
    {
      const int* src = bw_e + (size_t)v * 2 * E;
      const int* dst = src + E;
      zero_f32<<<1024, 256, 0, stream>>>(agg, (size_t)N * DIM + N);
      edge_agg<<<2048, 256, 0, stream>>>(inps, src, dst, agg, deg, E);
      gnn_gemm<<<ntiles, 32, 0, stream>>>(agg, deg,
                                          W_bw + (size_t)v * DIM * DHALF,
                                          b_bw + (size_t)v * DHALF, out, 0);
    }
    // forward direction -> cols [DHALF, DIM)
    {
      const int* src = fw_e + (size_t)v * 2 * E;
      const int* dst = src + E;
      zero_f32<<<1024, 256, 0, stream>>>(agg, (size_t)N * DIM + N);
      edge_agg<<<2048, 256, 0, stream>>>(inps, src, dst, agg, deg, E);
      gnn_gemm<<<ntiles, 32, 0, stream>>>(agg, deg,
                                          W_fw + (size_t)v * DIM * DHALF,
                                          b_fw + (size_t)v * DHALF, out, DHALF);
    }
  }

  final_fuse<<<ntiles, 32, 0, stream>>>(W1, b1, inps, out);
}